// ScaledDotProductAttention_4200478015739
// MI455X (gfx1250) — compile-verified
//
#include <hip/hip_runtime.h>

typedef __attribute__((ext_vector_type(2))) float v2f;
typedef __attribute__((ext_vector_type(4))) float v4f;
typedef __attribute__((ext_vector_type(8))) float v8f;

#define S_LEN   1024
#define DHEAD   64
#define SROW    1028              // padded LDS row stride in floats (4112 B, 16B-aligned)
#define SMEM_BYTES (16 * SROW * 4)
#define INV_T   0.125f            // 1 / TEMPERATURE
#define INV_DM1 (1.0f / 1023.0f)  // 1 / (dim - 1)

__global__ __launch_bounds__(32)
void rca_attn_kernel(const float* __restrict__ q,
                     const float* __restrict__ k,
                     const float* __restrict__ v,
                     float* __restrict__ out,
                     float* __restrict__ attn)
{
    extern __shared__ float smem[];   // 16 rows x SROW floats (scores -> attn)

    const int lane = threadIdx.x;     // 0..31 (wave32)
    const int m    = lane & 15;       // A-row / B-col / C-col index
    const int h    = lane >> 4;       // half-wave select
    const int qt   = blockIdx.x;      // query tile 0..63
    const int bh   = blockIdx.y;      // fused batch*head 0..127
    const int q0   = qt * 16;

    // ---------------- Phase 1: preload scaled Q A-fragments ----------------
    // A 16x4 f32 layout: lane holds row m; VGPR0/1 = K = k0+2h, k0+2h+1
    v2f qa[16];
    {
        const float* qrow = q + ((size_t)bh * S_LEN + q0 + m) * DHEAD;
        #pragma unroll
        for (int kk = 0; kk < 16; ++kk) {
            v2f t = *(const v2f*)(qrow + kk * 4 + 2 * h);
            qa[kk] = t * INV_T;
        }
    }

    // ---------------- Phase 2: scores = (Q/T) @ K^T via WMMA f32 -----------
    const float* kbase = k + (size_t)bh * S_LEN * DHEAD;
    for (int t = 0; t < 64; ++t) {
        const int n0 = t * 16;
        // B 4x16 layout: lane holds col n (= m); VGPR0/1 = K rows 2h, 2h+1
        const float* krow = kbase + (size_t)(n0 + m) * DHEAD;
        v8f c = {};
        #pragma unroll
        for (int kk = 0; kk < 16; ++kk) {
            v2f b = *(const v2f*)(krow + kk * 4 + 2 * h);
            c = __builtin_amdgcn_wmma_f32_16x16x4_f32(
                    false, qa[kk], false, b, (short)0, c, false, false);
        }
        // C/D layout: lane -> col n0+m; VGPR r -> row r + 8h
        #pragma unroll
        for (int r = 0; r < 8; ++r)
            smem[(r + 8 * h) * SROW + n0 + m] = c[r];
    }
    __syncthreads();

    // ---------------- Phase 3: softmax + reverse transform ------------------
    // All 32 lanes cooperate on each of the 16 rows (coalesced 512B stores).
    const size_t attnBase = ((size_t)bh * S_LEN + q0) * (size_t)S_LEN;
    for (int row = 0; row < 16; ++row) {
        float* srow = smem + row * SROW;
        // pass A: row max
        float mx = -3.402823466e38f;
        for (int j = lane; j < 256; j += 32) {
            v4f x = *(const v4f*)(srow + 4 * j);
            mx = fmaxf(mx, fmaxf(fmaxf(x.x, x.y), fmaxf(x.z, x.w)));
        }
        #pragma unroll
        for (int off = 16; off; off >>= 1)
            mx = fmaxf(mx, __shfl_xor(mx, off, 32));
        // pass B: exp + sum (store e back to LDS)
        float sum = 0.0f;
        for (int j = lane; j < 256; j += 32) {
            v4f x = *(const v4f*)(srow + 4 * j);
            v4f e;
            e.x = __expf(x.x - mx);
            e.y = __expf(x.y - mx);
            e.z = __expf(x.z - mx);
            e.w = __expf(x.w - mx);
            sum += (e.x + e.y) + (e.z + e.w);
            *(v4f*)(srow + 4 * j) = e;
        }
        #pragma unroll
        for (int off = 16; off; off >>= 1)
            sum += __shfl_xor(sum, off, 32);
        const float rinv = 1.0f / sum;
        // pass C: attn = (1 - p) / 1023; write LDS + streaming global store
        float* grow = attn + attnBase + (size_t)row * S_LEN;
        for (int j = lane; j < 256; j += 32) {
            v4f e = *(const v4f*)(srow + 4 * j);
            v4f a;
            a.x = (1.0f - e.x * rinv) * INV_DM1;
            a.y = (1.0f - e.y * rinv) * INV_DM1;
            a.z = (1.0f - e.z * rinv) * INV_DM1;
            a.w = (1.0f - e.w * rinv) * INV_DM1;
            *(v4f*)(srow + 4 * j) = a;
            __builtin_nontemporal_store(a, (v4f*)(grow + 4 * j));
        }
    }
    __syncthreads();

    // ---------------- Phase 4: out = attn @ V via WMMA f32 ------------------
    v8f acc0 = {}, acc1 = {}, acc2 = {}, acc3 = {};
    const float* vbase = v + (size_t)bh * S_LEN * DHEAD;
    #pragma unroll 2
    for (int k0 = 0; k0 < S_LEN; k0 += 4) {
        // A fragment from LDS: row m, K = k0+2h, k0+2h+1 (bank-conflict-free)
        v2f a = *(const v2f*)(smem + m * SROW + k0 + 2 * h);
        const float* vr0 = vbase + (size_t)(k0 + 2 * h) * DHEAD;
        const float* vr1 = vr0 + DHEAD;
        v2f b0 = { vr0[ 0 + m], vr1[ 0 + m] };
        v2f b1 = { vr0[16 + m], vr1[16 + m] };
        v2f b2 = { vr0[32 + m], vr1[32 + m] };
        v2f b3 = { vr0[48 + m], vr1[48 + m] };
        acc0 = __builtin_amdgcn_wmma_f32_16x16x4_f32(false, a, false, b0, (short)0, acc0, false, false);
        acc1 = __builtin_amdgcn_wmma_f32_16x16x4_f32(false, a, false, b1, (short)0, acc1, false, false);
        acc2 = __builtin_amdgcn_wmma_f32_16x16x4_f32(false, a, false, b2, (short)0, acc2, false, false);
        acc3 = __builtin_amdgcn_wmma_f32_16x16x4_f32(false, a, false, b3, (short)0, acc3, false, false);
    }
    float* obase = out + ((size_t)bh * S_LEN + q0) * DHEAD;
    #pragma unroll
    for (int r = 0; r < 8; ++r) {
        float* orow = obase + (size_t)(r + 8 * h) * DHEAD;
        orow[ 0 + m] = acc0[r];
        orow[16 + m] = acc1[r];
        orow[32 + m] = acc2[r];
        orow[48 + m] = acc3[r];
    }
}

extern "C" void kernel_launch(void* const* d_in, const int* in_sizes, int n_in,
                              void* d_out, int out_size, void* d_ws, size_t ws_size,
                              hipStream_t stream) {
    (void)in_sizes; (void)n_in; (void)d_ws; (void)ws_size; (void)out_size;
    const float* q = (const float*)d_in[0];
    const float* k = (const float*)d_in[1];
    const float* v = (const float*)d_in[2];
    float* out  = (float*)d_out;
    float* attn = out + (size_t)8 * 16 * 1024 * 64;   // out first, then attn

    dim3 grid(64, 128);   // (query tiles, batch*heads)
    dim3 block(32);       // one wave32 per workgroup
    rca_attn_kernel<<<grid, block, SMEM_BYTES, stream>>>(q, k, v, out, attn);
}